// VisibilityHeatmap_41841571398294
// MI455X (gfx1250) — compile-verified
//
#include <hip/hip_runtime.h>

// VisibilityHeatmap: per (b,k) gather heatmaps[b,k,v,u] (coords given as (u,v)),
// mask = in_bounds && val > 0.4. B=128, K=64, H=W=128. Output: 8192 floats (0/1).
//
// Latency-bound scattered gather (~1MB total HBM traffic @ 23.3TB/s => pure
// latency, zero matrix FLOPs; WMMA not applicable). CDNA5 paths used:
//  - async global->LDS DMA (global_load_async_to_lds_b64, ASYNCcnt) for coords
//  - non-temporal hint on the one-touch 512MB heatmap gather (protect 192MB L2)
//  - wave32: 64 blocks x 128 threads = 256 waves of independent loads in flight.

#define HM_H 128
#define HM_W 128
#define THRESH 0.4f
#define BLOCK 128

typedef int v2i __attribute__((ext_vector_type(2)));
typedef __attribute__((address_space(1))) v2i* gptr_v2i;
typedef __attribute__((address_space(3))) v2i* lptr_v2i;

__global__ __launch_bounds__(BLOCK) void VisibilityHeatmap_kernel(
    const int* __restrict__ coords,   // [N, 2] int32, (u, v)
    const float* __restrict__ heat,   // [N, H, W] float32
    float* __restrict__ out,          // [N] float32 (0.0 / 1.0)
    int n)
{
    __shared__ int s_coords[BLOCK * 2];

    const int tid = threadIdx.x;
    const int gid = blockIdx.x * BLOCK + tid;

#if defined(__gfx1250__) && __has_builtin(__builtin_amdgcn_global_load_async_to_lds_b64)
    // CDNA5 async global->LDS: DMA each lane's (u,v) pair straight into LDS,
    // no VGPR staging. Tracked by ASYNCcnt.
    if (gid < n) {
        __builtin_amdgcn_global_load_async_to_lds_b64(
            (gptr_v2i)(const_cast<int*>(coords) + (size_t)gid * 2),
            (lptr_v2i)(s_coords + tid * 2),
            /*offset=*/0, /*cpol=*/0);
    }
  #if __has_builtin(__builtin_amdgcn_s_wait_asynccnt)
    __builtin_amdgcn_s_wait_asynccnt(0);
  #else
    asm volatile("s_wait_asynccnt 0" ::: "memory");
  #endif
    __syncthreads();
#else
    // Fallback: coalesced b64 load through VGPRs.
    if (gid < n) {
        const int2 c = *(const int2*)(coords + (size_t)gid * 2);
        s_coords[tid * 2 + 0] = c.x;
        s_coords[tid * 2 + 1] = c.y;
    }
    __syncthreads();
#endif

    if (gid >= n) return;

    const int u = s_coords[tid * 2 + 0];   // coords[...,0] = u (col)
    const int v = s_coords[tid * 2 + 1];   // coords[...,1] = v (row)

    // (v > -1) & (u > -1) & (v < H) & (u < W)  ==  unsigned range checks
    const bool inb = ((unsigned)v < (unsigned)HM_H) && ((unsigned)u < (unsigned)HM_W);

    // clip for safe gather (matches reference: clip then mask)
    const int cv = v < 0 ? 0 : (v > HM_H - 1 ? HM_H - 1 : v);
    const int cu = u < 0 ? 0 : (u > HM_W - 1 ? HM_W - 1 : u);

    // one-touch scattered gather: non-temporal so the 512MB heatmap doesn't
    // evict useful L2 contents.
    const float* p = heat + (size_t)gid * (HM_H * HM_W) + (size_t)cv * HM_W + cu;
    const float val = __builtin_nontemporal_load(p);

    out[gid] = (inb && (val > THRESH)) ? 1.0f : 0.0f;
}

extern "C" void kernel_launch(void* const* d_in, const int* in_sizes, int n_in,
                              void* d_out, int out_size, void* d_ws, size_t ws_size,
                              hipStream_t stream) {
    const int*   coords = (const int*)d_in[0];    // [B,K,2] int32
    const float* heat   = (const float*)d_in[1];  // [B,K,H,W] float32
    float*       out    = (float*)d_out;          // [B,K] as float 0/1

    const int n = out_size;                       // B*K = 8192
    const int grid = (n + BLOCK - 1) / BLOCK;
    VisibilityHeatmap_kernel<<<grid, BLOCK, 0, stream>>>(coords, heat, out, n);
}